// CreateMultiScaleWarping_5961414607547
// MI455X (gfx1250) — compile-verified
//
#include <hip/hip_runtime.h>
#include <cstdint>

#define USE_ASYNC_LDS 1

static __device__ __forceinline__ void corner_of(int xi, int yi, int W, int H,
                                                 float wgt, int& sp, float& wv) {
  // zero-padding: invalid corners get weight 0; indices clamped for the load.
  int xc = xi < 0 ? 0 : (xi > W - 1 ? W - 1 : xi);
  int yc = yi < 0 ? 0 : (yi > H - 1 ? H - 1 : yi);
  bool valid = (xi >= 0) & (xi <= W - 1) & (yi >= 0) & (yi <= H - 1);
  sp = yc * W + xc;
  wv = valid ? wgt : 0.0f;
}

template <int H>
__global__ __launch_bounds__(256) void msw_warp_kernel(
    const float* __restrict__ img, const float* __restrict__ flow,
    float* __restrict__ out) {
  constexpr int C = 16;
  constexpr int W = H;
  constexpr unsigned HW = (unsigned)H * W;
  constexpr int LOG2W = (H == 512) ? 9 : (H == 256) ? 8 : (H == 128) ? 7 : 6;

  // Grid covers B*H*H exactly (multiple of 256): no bounds check, EXEC all-1s.
  const unsigned gid = blockIdx.x * 256u + threadIdx.x;
  const int w = (int)(gid & (W - 1));
  const int h = (int)((gid >> LOG2W) & (H - 1));
  const int b = (int)(gid >> (2 * LOG2W));

  float f0, f1;
#if USE_ASYNC_LDS
  // Stage this thread's flow pair via the CDNA5 async global->LDS path.
  __shared__ float sflow[2][256];
  {
    const float* g0 = flow + (unsigned)b * 2u * HW + (unsigned)h * W + w;
    const float* g1 = g0 + HW;
    unsigned l0 = (unsigned)(uintptr_t)&sflow[0][threadIdx.x];
    unsigned l1 = (unsigned)(uintptr_t)&sflow[1][threadIdx.x];
    asm volatile("global_load_async_to_lds_b32 %0, %1, off"
                 :: "v"(l0), "v"(g0) : "memory");
    asm volatile("global_load_async_to_lds_b32 %0, %1, off"
                 :: "v"(l1), "v"(g1) : "memory");
    asm volatile("s_wait_asynccnt 0" ::: "memory");
    f0 = sflow[0][threadIdx.x];
    f1 = sflow[1][threadIdx.x];
  }
#else
  {
    const float* g0 = flow + (unsigned)b * 2u * HW + (unsigned)h * W + w;
    f0 = g0[0];
    f1 = g0[HW];
  }
#endif

  // Faithful port of the reference coordinate math (incl. the (i-1) offset and
  // the row-based base grid feeding the x coordinate). All H-derived factors
  // are compile-time float literals now.
  constexpr float inv2 = 2.0f / (float)(H - 1);
  constexpr float invH2 = 2.0f / (float)H;
  constexpr float halfWm1 = 0.5f * (float)(W - 1);
  const float gx = (-1.0f + (float)(h - 1) * inv2) + f0 * invH2;
  const float gy = (-1.0f + (float)(w - 1) * inv2) + f1 * invH2;
  const float x = (gx + 1.0f) * halfWm1;
  const float y = (gy + 1.0f) * halfWm1;

  const float x0f = floorf(x), y0f = floorf(y);
  const int x0 = (int)x0f, y0 = (int)y0f;
  const float wx1 = x - x0f, wx0 = 1.0f - wx1;
  const float wy1 = y - y0f, wy0 = 1.0f - wy1;

  int sp00, sp10, sp01, sp11;
  float w00, w10, w01, w11;
  corner_of(x0,     y0,     W, H, wx0 * wy0, sp00, w00);
  corner_of(x0 + 1, y0,     W, H, wx1 * wy0, sp10, w10);
  corner_of(x0,     y0 + 1, W, H, wx0 * wy1, sp01, w01);
  corner_of(x0 + 1, y0 + 1, W, H, wx1 * wy1, sp11, w11);

  // Channel loop: 4 gathers + 1 NT store per plane, fully unrolled for MLP.
  const float* p = img + (size_t)((unsigned)b * C) * HW;
  float* o = out + (size_t)((unsigned)b * C) * HW + (unsigned)h * W + w;
#pragma unroll
  for (int c = 0; c < C; ++c) {
    float v = fmaf(w00, p[sp00],
              fmaf(w10, p[sp10],
              fmaf(w01, p[sp01], w11 * p[sp11])));
    __builtin_nontemporal_store(v, o);  // outputs never re-read: keep L2 for img
    p += HW;
    o += HW;
  }
}

extern "C" void kernel_launch(void* const* d_in, const int* in_sizes, int n_in,
                              void* d_out, int out_size, void* d_ws, size_t ws_size,
                              hipStream_t stream) {
  (void)in_sizes; (void)n_in; (void)out_size; (void)d_ws; (void)ws_size;
  const int B = 8, C = 16;
  float* out = (float*)d_out;
  size_t off = 0;
  for (int i = 0; i < 4; ++i) {
    const int H = 512 >> i;
    const float* img  = (const float*)d_in[2 * i + 0];
    const float* flow = (const float*)d_in[2 * i + 1];
    const int blocks = (int)(((long long)B * H * H) / 256);  // exact multiple
    switch (H) {
      case 512: msw_warp_kernel<512><<<blocks, 256, 0, stream>>>(img, flow, out + off); break;
      case 256: msw_warp_kernel<256><<<blocks, 256, 0, stream>>>(img, flow, out + off); break;
      case 128: msw_warp_kernel<128><<<blocks, 256, 0, stream>>>(img, flow, out + off); break;
      default:  msw_warp_kernel< 64><<<blocks, 256, 0, stream>>>(img, flow, out + off); break;
    }
    off += (size_t)B * C * H * H;
  }
}